// SelfSpace_33586644254896
// MI455X (gfx1250) — compile-verified
//
#include <hip/hip_runtime.h>
#include <hip/hip_bf16.h>

#define N_PTS 12288
#define DIMS  64
#define KNN   16
#define EPSV  1e-5f
#define DIAGB 1e6f
#define CHUNK 256
#define NCHUNK (N_PTS / CHUNK)   // 48
#define SENTINEL 3.0e38f

typedef __attribute__((ext_vector_type(2))) float v2f;
typedef __attribute__((ext_vector_type(8))) float v8f;

// ---------------- workspace layout (floats) ----------------
// ws[0..63]   : column sums s_d
// ws[64]      : S1 = sum_i ||x_i||^2
// ws[65]      : inv = 1/(sigma2 + EPS)
// ws[128+i]   : sq_i = ||x_i||^2, i in [0, N)

__global__ void zero_ws_kernel(float* __restrict__ ws) {
    if (threadIdx.x < 66) ws[threadIdx.x] = 0.0f;
}

__global__ __launch_bounds__(256) void row_sq_kernel(const float* __restrict__ coords,
                                                     float* __restrict__ ws) {
    const int i = blockIdx.x * 256 + threadIdx.x;          // exact: 48*256 = N
    const float4* r = (const float4*)(coords + (size_t)i * DIMS);
    float s = 0.0f;
#pragma unroll
    for (int q = 0; q < DIMS / 4; ++q) {
        float4 v = r[q];
        s += v.x * v.x + v.y * v.y + v.z * v.z + v.w * v.w;
    }
    ws[128 + i] = s;
    __shared__ float red[256];
    red[threadIdx.x] = s;
    __syncthreads();
    for (int off = 128; off > 0; off >>= 1) {
        if (threadIdx.x < off) red[threadIdx.x] += red[threadIdx.x + off];
        __syncthreads();
    }
    if (threadIdx.x == 0) atomicAdd(&ws[64], red[0]);
}

__global__ void col_sum_kernel(const float* __restrict__ coords,
                               float* __restrict__ ws) {
    const int d = threadIdx.x;           // 0..63
    const int row0 = blockIdx.x * 256;   // 48 blocks
    float s = 0.0f;
    for (int i = 0; i < 256; ++i)
        s += coords[(size_t)(row0 + i) * DIMS + d];
    atomicAdd(&ws[d], s);
}

__global__ void finalize_kernel(float* __restrict__ ws) {
    if (threadIdx.x == 0 && blockIdx.x == 0) {
        double s2 = 0.0;
        for (int d = 0; d < DIMS; ++d) s2 += (double)ws[d] * (double)ws[d];
        const double n = (double)N_PTS;
        // sum_ij dist2 = 2N*S1 - 2*||sum x||^2 + N^2*eps + N*diag_bias
        double sumAll = 2.0 * n * (double)ws[64] - 2.0 * s2 + n * n * (double)EPSV + n * (double)DIAGB;
        double sigma2 = sumAll / (n * n);
        ws[65] = (float)(1.0 / (sigma2 + (double)EPSV));
    }
}

// Register-resident running top-K (smallest). thresh = current max of kept set.
__device__ __forceinline__ void topk_insert(float (&vals)[KNN], int (&idx)[KNN],
                                            float cand, int cidx, float& thresh) {
    if (cand < thresh) {
        float mx = vals[0];
        int mp = 0;
#pragma unroll
        for (int e = 1; e < KNN; ++e) {
            if (vals[e] > mx) { mx = vals[e]; mp = e; }
        }
#pragma unroll
        for (int e = 0; e < KNN; ++e) {
            if (e == mp) { vals[e] = cand; idx[e] = cidx; }
        }
        mx = vals[0];
#pragma unroll
        for (int e = 1; e < KNN; ++e) mx = fmaxf(mx, vals[e]);
        thresh = mx;
    }
}

__global__ __launch_bounds__(256) void knn_lap_kernel(const float* __restrict__ coords,
                                                      const float* __restrict__ pot,
                                                      const float* __restrict__ ws,
                                                      float* __restrict__ out) {
    __shared__ float lds_d2[16][CHUNK];        // 16 KB: dist2 tile for this chunk
    __shared__ float lds_cval[16][16][KNN];    // 16 KB: per-(row,subthread) candidates
    __shared__ int   lds_cidx[16][16][KNN];    // 16 KB
    __shared__ float lds_sqi[16];
    __shared__ float lds_vi[16];

    const float* __restrict__ sq = ws + 128;
    const int tid  = threadIdx.x;
    const int lane = tid & 31;
    const int wave = tid >> 5;                 // 8 waves
    const int i0   = blockIdx.x * 16;
    const int m    = lane & 15;                // M (A) / N (B,C) index
    const int hi   = lane >> 4;

    if (tid < 16) {
        lds_sqi[tid] = sq[i0 + tid];
        lds_vi[tid]  = pot[i0 + tid];
    }

    // --- Preload A fragments for this 16-row block (resident in VGPRs) ---
    // ISA f32 A 16x4 layout: lane m = row; VGPR0 = {K, K+2}, VGPR1 = {K+1, K+3} split by hi.
    v2f a[16];
    {
        const float* arow = coords + (size_t)(i0 + m) * DIMS;
#pragma unroll
        for (int kk = 0; kk < 16; ++kk) {
            a[kk].x = arow[4 * kk + 2 * hi];
            a[kk].y = arow[4 * kk + 2 * hi + 1];
        }
    }

    // per-thread running top-16
    float tv[KNN];
    int   tix[KNN];
#pragma unroll
    for (int e = 0; e < KNN; ++e) { tv[e] = SENTINEL; tix[e] = 0; }
    float thresh = SENTINEL;

    const int selrow = tid >> 4;   // 16 threads per row
    const int selsub = tid & 15;

    __syncthreads();

    for (int cc = 0; cc < NCHUNK; ++cc) {
        const int j0c = cc * CHUNK;
#pragma unroll
        for (int t = 0; t < 2; ++t) {
            const int tile = wave * 2 + t;           // 16 tiles cover 256 columns
            const int j0   = j0c + tile * 16;
            const float* brow = coords + (size_t)(j0 + m) * DIMS;
            if (cc + 1 < NCHUNK) {
                // hint next chunk's column block toward L2
                __builtin_prefetch(coords + (size_t)(j0 + CHUNK + m) * DIMS, 0, 1);
            }
            v8f acc = {};
#pragma unroll
            for (int kk = 0; kk < 16; ++kk) {
                v2f b;
                b.x = brow[4 * kk + 2 * hi];
                b.y = brow[4 * kk + 2 * hi + 1];
                // D = A x B + C  (16x16x4 f32 WMMA, chained over K=64)
                acc = __builtin_amdgcn_wmma_f32_16x16x4_f32(
                    false, a[kk], false, b, (short)0, acc, false, false);
            }
            const float sqj  = sq[j0 + m];
            const int   gcol = j0 + m;
#pragma unroll
            for (int r = 0; r < 8; ++r) {
                const int row = r + 8 * hi;          // ISA C/D layout
                float d2 = lds_sqi[row] + sqj - 2.0f * acc[r] + EPSV;
                if (i0 + row == gcol) d2 += DIAGB;   // exclude self
                lds_d2[row][tile * 16 + m] = d2;
            }
        }
        __syncthreads();
        // streaming selection: 16 threads per row, strided columns
#pragma unroll
        for (int q = 0; q < 16; ++q) {
            const int c = selsub + 16 * q;
            topk_insert(tv, tix, lds_d2[selrow][c], j0c + c, thresh);
        }
        __syncthreads();
    }

    // dump per-thread candidate lists
#pragma unroll
    for (int e = 0; e < KNN; ++e) {
        lds_cval[selrow][selsub][e] = tv[e];
        lds_cidx[selrow][selsub][e] = tix[e];
    }
    __syncthreads();

    // one thread per row: merge 256 candidates -> final 16, compute Laplacian
    if (selsub == 0) {
        float fv[KNN];
        int   fx[KNN];
#pragma unroll
        for (int e = 0; e < KNN; ++e) { fv[e] = SENTINEL; fx[e] = 0; }
        float th2 = SENTINEL;
        for (int s = 0; s < 16; ++s) {
#pragma unroll
            for (int e = 0; e < KNN; ++e)
                topk_insert(fv, fx, lds_cval[selrow][s][e], lds_cidx[selrow][s][e], th2);
        }
        const float inv = ws[65];
        const float vi  = lds_vi[selrow];
        float lap = 0.0f;
#pragma unroll
        for (int e = 0; e < KNN; ++e) {
            const float w = __expf(-fv[e] * inv);    // exp(knn_val/(sigma2+eps)), knn_val = -d2
            lap += w * (pot[fx[e]] - vi);
        }
        out[i0 + selrow] = lap;
    }
}

extern "C" void kernel_launch(void* const* d_in, const int* in_sizes, int n_in,
                              void* d_out, int out_size, void* d_ws, size_t ws_size,
                              hipStream_t stream) {
    const float* coords = (const float*)d_in[0];
    const float* pot    = (const float*)d_in[1];
    // d_in[2] is k (==16), hardcoded as KNN
    float* out = (float*)d_out;
    float* ws  = (float*)d_ws;

    zero_ws_kernel<<<1, 128, 0, stream>>>(ws);
    row_sq_kernel<<<N_PTS / 256, 256, 0, stream>>>(coords, ws);
    col_sum_kernel<<<N_PTS / 256, DIMS, 0, stream>>>(coords, ws);
    finalize_kernel<<<1, 64, 0, stream>>>(ws);
    knn_lap_kernel<<<N_PTS / 16, 256, 0, stream>>>(coords, pot, ws, out);
}